// PCSA_10368051052974
// MI455X (gfx1250) — compile-verified
//
#include <hip/hip_runtime.h>

// PCSA fused pipeline for MI455X (gfx1250), wave32, f32 WMMA path.
// h = silu(X@Wd^T+bd); group-sorted U^T x -> LN -> silu adapter residual
// -> U f; combine with gather; out = h'@Wu^T+bu. Dead branch-0 skipped.
// CDNA5 paths: v_wmma_f32_16x16x4_f32 for all GEMMs, TDM tensor_load_to_lds
// (with D# LDS padding = bank-conflict pad) for the U tile, and
// global_load_async_to_lds_b128 for the gathered sub_x rows.

typedef float v2f __attribute__((ext_vector_type(2)));
typedef float v8f __attribute__((ext_vector_type(8)));
typedef unsigned int v4u __attribute__((ext_vector_type(4)));
typedef int v4i __attribute__((ext_vector_type(4)));
typedef int v8i __attribute__((ext_vector_type(8)));

#define B_   16
#define N_   8192
#define DIM_ 384
#define R_   64
#define G_   128
#define S_   64
#define BN_  (B_ * N_)

__device__ __forceinline__ float silu(float x) {
    return x / (1.0f + __expf(-x));
}

// D = A(16x4) x B(4x16) + C, f32, wave32.
__device__ __forceinline__ v8f wmma4(v2f a, v2f b, v8f c) {
    return __builtin_amdgcn_wmma_f32_16x16x4_f32(false, a, false, b,
                                                 (short)0, c, false, false);
}

// ---------------------------------------------------------------------------
// Kernel 1: H[m, r] = silu( X[m,:] . Wd[r,:] + bd[r] ),  m in [0, B*N)
// One wave owns one 16-row tile and all 4 rank tiles -> X read exactly once.
// ---------------------------------------------------------------------------
__global__ __launch_bounds__(256) void k_down_silu(
    const float* __restrict__ X, const float* __restrict__ Wd,
    const float* __restrict__ bd, float* __restrict__ H) {
    const int tid  = threadIdx.x;
    const int wave = tid >> 5;
    const int lane = tid & 31;
    const int lr   = lane & 15;
    const int half = lane >> 4;

    const size_t row0 = (size_t)blockIdx.x * 128 + (size_t)wave * 16;
    const float* arow = X + (row0 + (size_t)lr) * DIM_;

    v8f zero = {};
    v8f acc[4] = {zero, zero, zero, zero};

    #pragma unroll 4
    for (int k0 = 0; k0 < DIM_; k0 += 4) {
        const int ka = k0 + 2 * half;
        v2f a;
        a.x = arow[ka];
        a.y = arow[ka + 1];
        #pragma unroll
        for (int nt = 0; nt < 4; ++nt) {
            // B[k][n] = Wd^T[k][n] = Wd[n][k]
            const float* brow = Wd + (size_t)(nt * 16 + lr) * DIM_;
            v2f b;
            b.x = brow[ka];
            b.y = brow[ka + 1];
            acc[nt] = wmma4(a, b, acc[nt]);
        }
    }

    #pragma unroll
    for (int nt = 0; nt < 4; ++nt) {
        const int col  = nt * 16 + lr;
        const float bv = bd[col];
        #pragma unroll
        for (int i = 0; i < 8; ++i) {
            const size_t row = row0 + i + 8 * half;
            H[row * R_ + col] = silu(acc[nt][i] + bv);
        }
    }
}

// ---------------------------------------------------------------------------
// Kernel 2: per-(b,g) fused branch-1:
//   X = gather(H, idx0) ; F = U^T X ; F += silu(LN(F)@Wa^T + ba) ; Y = U F
// One block per group, 8 waves. U tile comes in via TDM (descriptor pads LDS
// rows to 65 floats -> conflict-free column reads); gathered X rows come in
// via per-lane async global->LDS B128 copies.
// ---------------------------------------------------------------------------
__global__ __launch_bounds__(256) void k_group(
    const float* __restrict__ subU1, const long long* __restrict__ idx0,
    const float* __restrict__ H, const float* __restrict__ Wa,
    const float* __restrict__ ba, const float* __restrict__ gamma,
    const float* __restrict__ beta, float* __restrict__ Y) {
    __shared__ float Us[S_][S_ + 1];
    __shared__ float Xs[S_][S_ + 1];   // sub_x, later reused as LN(F)
    __shared__ float Fs[S_][S_];

    const int tid = threadIdx.x;
    const int b   = blockIdx.x >> 7;   // / G_
    const int gi  = blockIdx.x & (G_ - 1);
    const float* U = subU1 + (size_t)blockIdx.x * (S_ * S_);

    const int wave = tid >> 5;
    const int lane = tid & 31;
    const int lr   = lane & 15;
    const int half = lane >> 4;

    // ---- Stage U via Tensor Data Mover: 64x64 f32 contiguous tile, LDS rows
    // padded to 65 floats by D# pad (interval = 64 DWORDs, amount = 1 DWORD).
    if (wave == 0) {
        const unsigned lds_u = (unsigned)(unsigned long long)&Us[0][0];
        const unsigned long long ga = (unsigned long long)U;
        v4u g0;
        g0.x = 1u;                                              // count = 1
        g0.y = lds_u;                                           // lds_addr
        g0.z = (unsigned)ga;                                    // global_addr lo
        g0.w = (unsigned)((ga >> 32) & 0x01FFFFFFu) | (2u << 30); // hi | type=2
        v8i g1;
        g1[0] = (int)((2u << 16) |        // data_size = 4 bytes
                      (1u << 20) |        // pad_enable
                      (5u << 22));        // pad_interval: 64 DWORDs (amount=1)
        g1[1] = (int)(64u << 16);         // tensor_dim0 = 64
        g1[2] = (int)(64u << 16);         // tensor_dim1 = 64
        g1[3] = (int)(64u << 16);         // tile_dim0 = 64
        g1[4] = 64;                       // tile_dim1 = 64
        g1[5] = 64;                       // tensor_dim0_stride = 64
        g1[6] = 0;
        g1[7] = 0;
        v4i gz = {0, 0, 0, 0};
#if __has_include(<hip/amd_detail/amd_gfx1250_TDM.h>)
        v8i gz8 = {0, 0, 0, 0, 0, 0, 0, 0};
        __builtin_amdgcn_tensor_load_to_lds(g0, g1, gz, gz, gz8, 0);
#else
        __builtin_amdgcn_tensor_load_to_lds(g0, g1, gz, gz, 0);
#endif
        __builtin_amdgcn_s_wait_tensorcnt(0);
    }

    // ---- Stage gathered X rows via async global->LDS copies (ASYNCcnt).
    {
        const unsigned lds_x = (unsigned)(unsigned long long)&Xs[0][0];
        #pragma unroll
        for (int j = 0; j < 4; ++j) {
            const int q  = tid + 256 * j;   // float4 chunk id, [0, 1024)
            const int t  = q >> 4;          // source row within group
            const int rq = (q & 15) * 4;    // starting rank column
            const long long src = idx0[(size_t)b * N_ + gi * S_ + t];
            const unsigned long long ga = (unsigned long long)(
                H + ((size_t)b * N_ + (size_t)src) * R_ + rq);
            const unsigned la = lds_x + (unsigned)(t * (S_ + 1) + rq) * 4u;
            asm volatile("global_load_async_to_lds_b128 %0, %1, off"
                         :: "v"(la), "v"(ga) : "memory");
        }
        asm volatile("s_wait_asynccnt 0x0" ::: "memory");
    }
    __syncthreads();

    const v8f zero = {};

    // GEMM1: F[s][r] = sum_t U[t][s] * X[t][r]   (A = U^T)
    for (int t = wave; t < 16; t += 8) {
        const int m0 = (t >> 2) * 16, n0 = (t & 3) * 16;
        v8f c = zero;
        #pragma unroll
        for (int k0 = 0; k0 < S_; k0 += 4) {
            const int ka = k0 + 2 * half;
            v2f a, bb;
            a.x  = Us[ka][m0 + lr];      // row-contiguous LDS read
            a.y  = Us[ka + 1][m0 + lr];
            bb.x = Xs[ka][n0 + lr];
            bb.y = Xs[ka + 1][n0 + lr];
            c = wmma4(a, bb, c);
        }
        #pragma unroll
        for (int i = 0; i < 8; ++i)
            Fs[m0 + i + 8 * half][n0 + lr] = c[i];
    }
    __syncthreads();

    // LayerNorm over rank dim (64), result into Xs (G)
    if (tid < S_) {
        const int row = tid;
        float m = 0.f, v = 0.f;
        #pragma unroll 8
        for (int j = 0; j < S_; ++j) {
            const float x = Fs[row][j];
            m += x;
            v += x * x;
        }
        m *= (1.0f / S_);
        v = v * (1.0f / S_) - m * m;
        const float inv = rsqrtf(v + 1e-5f);
        #pragma unroll 8
        for (int j = 0; j < S_; ++j)
            Xs[row][j] = (Fs[row][j] - m) * inv * gamma[j] + beta[j];
    }
    __syncthreads();

    // GEMM2: adapt = G @ Wa^T ; F += silu(adapt + ba)  (per-wave disjoint F tiles)
    for (int t = wave; t < 16; t += 8) {
        const int m0 = (t >> 2) * 16, n0 = (t & 3) * 16;
        v8f c = zero;
        #pragma unroll
        for (int k0 = 0; k0 < S_; k0 += 4) {
            const int ka = k0 + 2 * half;
            v2f a, bb;
            a.x  = Xs[m0 + lr][ka];       // stride-65 column read: conflict-free
            a.y  = Xs[m0 + lr][ka + 1];
            bb.x = Wa[(size_t)(n0 + lr) * R_ + ka];  // B[k][n] = Wa[n][k]
            bb.y = Wa[(size_t)(n0 + lr) * R_ + ka + 1];
            c = wmma4(a, bb, c);
        }
        const int col  = n0 + lr;
        const float bv = ba[col];
        #pragma unroll
        for (int i = 0; i < 8; ++i)
            Fs[m0 + i + 8 * half][col] += silu(c[i] + bv);
    }
    __syncthreads();

    // GEMM3: Y[s][r] = sum_t U[s][t] * F[t][r]
    for (int t = wave; t < 16; t += 8) {
        const int m0 = (t >> 2) * 16, n0 = (t & 3) * 16;
        v8f c = zero;
        #pragma unroll
        for (int k0 = 0; k0 < S_; k0 += 4) {
            const int ka = k0 + 2 * half;
            v2f a, bb;
            a.x  = Us[m0 + lr][ka];       // stride-65 column read
            a.y  = Us[m0 + lr][ka + 1];
            bb.x = Fs[ka][n0 + lr];
            bb.y = Fs[ka + 1][n0 + lr];
            c = wmma4(a, bb, c);
        }
        #pragma unroll
        for (int i = 0; i < 8; ++i) {
            const int srow = m0 + i + 8 * half;
            Y[((size_t)b * N_ + gi * S_ + srow) * R_ + n0 + lr] = c[i];
        }
    }
}

// ---------------------------------------------------------------------------
// Kernel 3: H[b,n,:] += Y[b,n,:] + Y[b, idx1[b,n], :]   (float4, in-place)
// ---------------------------------------------------------------------------
__global__ __launch_bounds__(256) void k_combine(
    float* __restrict__ H, const float* __restrict__ Y,
    const long long* __restrict__ idx1) {
    const size_t q = (size_t)blockIdx.x * 256 + threadIdx.x;  // quad index
    const size_t m = q >> 4;       // row in [0, B*N)
    const int   qc = (int)(q & 15);
    const int   b  = (int)(m >> 13);      // / N_
    const int   n  = (int)(m & (N_ - 1));

    const long long src = idx1[(size_t)b * N_ + n];
    const float4* Y4 = (const float4*)Y;
    float4*       H4 = (float4*)H;

    float4 hv = H4[q];
    float4 ya = Y4[q];
    float4 yb = Y4[(((size_t)b * N_ + (size_t)src) << 4) + qc];
    hv.x += ya.x + yb.x;
    hv.y += ya.y + yb.y;
    hv.z += ya.z + yb.z;
    hv.w += ya.w + yb.w;
    H4[q] = hv;
}

// ---------------------------------------------------------------------------
// Kernel 4: out[m, d] = H[m,:] . Wu[d,:] + bu[d]
// One wave per 16-row tile x 6 column tiles (nb selects which 6 of 24).
// ---------------------------------------------------------------------------
__global__ __launch_bounds__(256) void k_up(
    const float* __restrict__ H, const float* __restrict__ Wu,
    const float* __restrict__ bu, float* __restrict__ out) {
    const int tid  = threadIdx.x;
    const int wave = tid >> 5;
    const int lane = tid & 31;
    const int lr   = lane & 15;
    const int half = lane >> 4;

    const int task  = blockIdx.x * 8 + wave;   // 32768 tasks
    const size_t mt = (size_t)(task >> 2);
    const int nb    = task & 3;

    const size_t row0 = mt * 16;
    const float* arow = H + (row0 + (size_t)lr) * R_;

    v8f zero = {};
    v8f acc[6] = {zero, zero, zero, zero, zero, zero};

    #pragma unroll 4
    for (int k0 = 0; k0 < R_; k0 += 4) {
        const int ka = k0 + 2 * half;
        v2f a;
        a.x = arow[ka];
        a.y = arow[ka + 1];
        #pragma unroll
        for (int j = 0; j < 6; ++j) {
            const int col = (nb * 6 + j) * 16 + lr;  // B[k][n] = Wu[n][k]
            v2f b;
            b.x = Wu[(size_t)col * R_ + ka];
            b.y = Wu[(size_t)col * R_ + ka + 1];
            acc[j] = wmma4(a, b, acc[j]);
        }
    }

    #pragma unroll
    for (int j = 0; j < 6; ++j) {
        const int col  = (nb * 6 + j) * 16 + lr;
        const float bv = bu[col];
        #pragma unroll
        for (int i = 0; i < 8; ++i) {
            const size_t row = row0 + i + 8 * half;
            out[row * DIM_ + col] = acc[j][i] + bv;
        }
    }
}

extern "C" void kernel_launch(void* const* d_in, const int* in_sizes, int n_in,
                              void* d_out, int out_size, void* d_ws, size_t ws_size,
                              hipStream_t stream) {
    const float*     input = (const float*)d_in[0];
    const float*     subU  = (const float*)d_in[1];
    const long long* idx   = (const long long*)d_in[2];
    const float*     Wd    = (const float*)d_in[3];
    const float*     bd    = (const float*)d_in[4];
    const float*     Wu    = (const float*)d_in[5];
    const float*     bu    = (const float*)d_in[6];
    const float*     Wa    = (const float*)d_in[7];
    const float*     ba    = (const float*)d_in[8];
    const float*     gamma = (const float*)d_in[9];
    const float*     beta  = (const float*)d_in[10];

    float* H = (float*)d_ws;                 // [B*N, 64]  33.5 MB
    float* Y = H + (size_t)BN_ * R_;         // [B*N, 64]  33.5 MB
    float* out = (float*)d_out;

    const float*     subU1 = subU + (size_t)B_ * G_ * S_ * S_;  // branch 1
    const long long* idx0  = idx;
    const long long* idx1  = idx + (size_t)B_ * N_;

    k_down_silu<<<BN_ / 128, 256, 0, stream>>>(input, Wd, bd, H);
    k_group<<<B_ * G_, 256, 0, stream>>>(subU1, idx0, H, Wa, ba, gamma, beta, Y);
    k_combine<<<(BN_ * 16) / 256, 256, 0, stream>>>(H, Y, idx1);
    k_up<<<((BN_ / 16) * 4) / 8, 256, 0, stream>>>(H, Wu, bu, out);
}